// Encoder_63213328663163
// MI455X (gfx1250) — compile-verified
//
#include <hip/hip_runtime.h>
#include <hip/hip_bf16.h>
#include <math.h>

typedef __attribute__((ext_vector_type(2))) float v2f;
typedef __attribute__((ext_vector_type(8))) float v8f;

#define BB 16
#define DM 128
#define HD 8
#define DK 16
#define NN 1024
#define LL 12

__device__ __forceinline__ v8f wmma4(v2f a, v2f b, v8f c) {
  // D(16x16,f32) = A(16x4,f32) x B(4x16,f32) + C   -- native fp32 WMMA on gfx1250
  return __builtin_amdgcn_wmma_f32_16x16x4_f32(false, a, false, b, (short)0, c,
                                               false, false);
}

// ---------------------------------------------------------------------------
// Transpose (B,D,N,L) -> [b][l][d][n]  (n contiguous) via LDS staging.
// ---------------------------------------------------------------------------
__global__ __launch_bounds__(256) void k_transpose_in(const float* __restrict__ in,
                                                      float* __restrict__ Xt) {
  __shared__ float tile[LL * NN];          // tile[n*12 + l]
  int b = blockIdx.x >> 7;
  int d = blockIdx.x & 127;
  const float* src = in + (size_t)(b * DM + d) * (NN * LL);
  for (int o = threadIdx.x; o < NN * LL; o += 256) tile[o] = src[o];
  __syncthreads();
  for (int o = threadIdx.x; o < NN * LL; o += 256) {
    int l = o >> 10, n = o & 1023;
    Xt[((size_t)(b * LL + l) * DM + d) * NN + n] = tile[n * LL + l];
  }
}

// [b][l][d][n] -> (B,D,N,L)
__global__ __launch_bounds__(256) void k_transpose_out(const float* __restrict__ U,
                                                       float* __restrict__ out) {
  __shared__ float tile[LL * NN];          // tile[l*1024 + n]
  int b = blockIdx.x >> 7;
  int d = blockIdx.x & 127;
  for (int o = threadIdx.x; o < NN * LL; o += 256) {
    int l = o >> 10, n = o & 1023;
    tile[o] = U[((size_t)(b * LL + l) * DM + d) * NN + n];
  }
  __syncthreads();
  float* dst = out + (size_t)(b * DM + d) * (NN * LL);
  for (int o = threadIdx.x; o < NN * LL; o += 256) {
    int n = o / LL, l = o - n * LL;
    dst[o] = tile[l * NN + n];
  }
}

// ---------------------------------------------------------------------------
// key = softmax(memory * 1/sqrt(d_k), axis=-1); one thread per (h,l,n) row.
// ---------------------------------------------------------------------------
__global__ __launch_bounds__(256) void k_key(const float* __restrict__ mem,
                                             float* __restrict__ keysm) {
  int r = blockIdx.x * 256 + threadIdx.x;  // 8*12*1024 = 98304 rows
  const float* p = mem + (size_t)r * DK;
  float v[DK];
  float mx = -1e30f;
  for (int i = 0; i < DK; ++i) { v[i] = p[i] * 0.25f; mx = fmaxf(mx, v[i]); }
  float s = 0.f;
  for (int i = 0; i < DK; ++i) { v[i] = __expf(v[i] - mx); s += v[i]; }
  float inv = 1.f / s;
  float* q = keysm + (size_t)r * DK;
  for (int i = 0; i < DK; ++i) q[i] = v[i] * inv;
}

// ---------------------------------------------------------------------------
// Adaptive part. Per node n: row of Aapt = softmax(relu(nv1[n,:] @ nv2)),
// bias[n,:] = row @ b_pool, sw[n] = rowsum(row) * sum(w_pool).
// ---------------------------------------------------------------------------
__global__ __launch_bounds__(256) void k_apt(const float* __restrict__ nv1,
                                             const float* __restrict__ nv2,
                                             const float* __restrict__ w_pool,
                                             const float* __restrict__ b_pool,
                                             float* __restrict__ biasg,
                                             float* __restrict__ swg) {
  __shared__ float a[NN];
  __shared__ float red[256];
  __shared__ float r1[16];
  int n = blockIdx.x, tid = threadIdx.x;
  if (tid < 10) r1[tid] = nv1[n * 10 + tid];
  __syncthreads();
  for (int m = tid; m < NN; m += 256) {
    float s = 0.f;
    for (int j = 0; j < 10; ++j) s += r1[j] * nv2[j * NN + m];
    a[m] = fmaxf(s, 0.f);                      // relu before softmax
  }
  __syncthreads();
  float mx = -1e30f;
  for (int m = tid; m < NN; m += 256) mx = fmaxf(mx, a[m]);
  red[tid] = mx; __syncthreads();
  for (int s = 128; s > 0; s >>= 1) {
    if (tid < s) red[tid] = fmaxf(red[tid], red[tid + s]);
    __syncthreads();
  }
  mx = red[0]; __syncthreads();
  float ps = 0.f;
  for (int m = tid; m < NN; m += 256) { float e = __expf(a[m] - mx); a[m] = e; ps += e; }
  red[tid] = ps; __syncthreads();
  for (int s = 128; s > 0; s >>= 1) {
    if (tid < s) red[tid] += red[tid + s];
    __syncthreads();
  }
  float S = red[0], invS = 1.f / S;
  __syncthreads();
  if (tid < DK) {
    float bs = 0.f;
    for (int m = 0; m < NN; ++m) bs += a[m] * b_pool[m * DK + tid];
    biasg[n * DK + tid] = bs * invS;
  }
  if (tid == 0) {
    float wsum = 0.f;
    for (int i = 0; i < LL * 9; ++i) wsum += w_pool[i];
    float rs = 0.f;
    for (int m = 0; m < NN; ++m) rs += a[m] * invS;  // == 1 numerically, faithful
    swg[n] = rs * wsum;
  }
}

// ---------------------------------------------------------------------------
// Fused attention per (b,h,l). 8 waves, 256 threads.
//   q,v = relu(W @ X + b)     (2 n-tiles jointly: 4 WMMA per shared A/B fetch)
//   query = softmax(q*scale)
//   kv = key^T @ v            (per-wave WMMA partials + LDS reduction)
//   attn = query @ kv + v*sw[n] + bias[n,:]
// ---------------------------------------------------------------------------
__global__ __launch_bounds__(256) void k_attn(
    const float* __restrict__ Xt, const float* __restrict__ qW,
    const float* __restrict__ qb, const float* __restrict__ vW,
    const float* __restrict__ vb, const float* __restrict__ keysm,
    const float* __restrict__ biasg, const float* __restrict__ swg,
    float* __restrict__ T1) {
  __shared__ float qWl[DK * DM];
  __shared__ float vWl[DK * DM];
  __shared__ float qls[DK * NN];   // q -> query -> output staging ([row][n])
  __shared__ float vls[DK * NN];   // v ([y][n])
  __shared__ float kvpart[8 * 256];
  __shared__ float kvl[DK * DK];
  __shared__ float qbl[DK], vbl[DK];

  int blk = blockIdx.x;
  int l = blk % LL, h = (blk / LL) % HD, b = blk / (LL * HD);
  int tid = threadIdx.x;
  int lane = tid & 31, lo = lane & 15, hi = lane >> 4, wv = tid >> 5;

  for (int i = tid; i < DK * DM; i += 256) {
    int o = i / DM, c = i % DM;
    qWl[i] = qW[(h * DK + o) * DM + c];
    vWl[i] = vW[(h * DK + o) * DM + c];
  }
  if (tid < DK) { qbl[tid] = qb[h * DK + tid]; vbl[tid] = vb[h * DK + tid]; }
  __syncthreads();

  const float* S = Xt + (size_t)(b * LL + l) * DM * NN;

  // ---- q,v GEMMs: two 16(ch) x 16(n) tiles per iteration, K = 128 ----
  for (int tp = wv; tp < 32; tp += 8) {
    int n0 = tp * 32;
    v8f aq0 = {}, av0 = {}, aq1 = {}, av1 = {};
    for (int kk = 0; kk < 32; ++kk) {
      int kr = kk * 4 + hi * 2;
      v2f Aq, Av, B0, B1;
      Aq.x = qWl[lo * DM + kr];     Aq.y = qWl[lo * DM + kr + 1];
      Av.x = vWl[lo * DM + kr];     Av.y = vWl[lo * DM + kr + 1];
      B0.x = S[(kr + 0) * NN + n0 + lo];
      B0.y = S[(kr + 1) * NN + n0 + lo];
      B1.x = S[(kr + 0) * NN + n0 + 16 + lo];
      B1.y = S[(kr + 1) * NN + n0 + 16 + lo];
      aq0 = wmma4(Aq, B0, aq0);
      av0 = wmma4(Av, B0, av0);
      aq1 = wmma4(Aq, B1, aq1);
      av1 = wmma4(Av, B1, av1);
    }
#pragma unroll
    for (int r = 0; r < 8; ++r) {
      int x = r + 8 * hi;
      qls[x * NN + n0 + lo]      = fmaxf(aq0[r] + qbl[x], 0.f);
      vls[x * NN + n0 + lo]      = fmaxf(av0[r] + vbl[x], 0.f);
      qls[x * NN + n0 + 16 + lo] = fmaxf(aq1[r] + qbl[x], 0.f);
      vls[x * NN + n0 + 16 + lo] = fmaxf(av1[r] + vbl[x], 0.f);
    }
  }
  __syncthreads();

  // ---- query softmax over channel dim per column n ----
  for (int n = tid; n < NN; n += 256) {
    float vbuf[DK], mx = -1e30f;
    for (int x = 0; x < DK; ++x) { vbuf[x] = qls[x * NN + n] * 0.25f; mx = fmaxf(mx, vbuf[x]); }
    float s = 0.f;
    for (int x = 0; x < DK; ++x) { vbuf[x] = __expf(vbuf[x] - mx); s += vbuf[x]; }
    float inv = 1.f / s;
    for (int x = 0; x < DK; ++x) qls[x * NN + n] = vbuf[x] * inv;
  }
  __syncthreads();

  // ---- kv[x][y] = sum_n key[n,x]*v[n,y]; per-wave partial over 128 n ----
  const float* KS = keysm + (size_t)(h * LL + l) * NN * DK;
  v8f kvacc = {};
  for (int kk = 0; kk < 32; ++kk) {
    int n0 = wv * 128 + kk * 4 + hi * 2;
    v2f A, Bv;
    A.x = KS[(n0 + 0) * DK + lo];
    A.y = KS[(n0 + 1) * DK + lo];
    Bv.x = vls[lo * NN + n0 + 0];
    Bv.y = vls[lo * NN + n0 + 1];
    kvacc = wmma4(A, Bv, kvacc);
  }
#pragma unroll
  for (int r = 0; r < 8; ++r) kvpart[wv * 256 + r * 32 + lane] = kvacc[r];
  __syncthreads();
  {
    float s = 0.f;
    for (int w = 0; w < 8; ++w) s += kvpart[w * 256 + tid];
    int r = tid >> 5, ln = tid & 31;
    int x = r + 8 * (ln >> 4), y = ln & 15;
    kvl[x * DK + y] = s;
  }
  __syncthreads();

  // ---- attn tile 16(n) x 16(y): query(n,x) @ kv(x,y) + dyn; stage in qls ----
  for (int t = wv; t < 64; t += 8) {
    int n0 = t * 16;
    v8f acc = {};
    for (int k0 = 0; k0 < DK; k0 += 4) {
      int k = k0 + hi * 2;
      v2f A, Bk;
      A.x = qls[(k + 0) * NN + n0 + lo];
      A.y = qls[(k + 1) * NN + n0 + lo];
      Bk.x = kvl[(k + 0) * DK + lo];
      Bk.y = kvl[(k + 1) * DK + lo];
      acc = wmma4(A, Bk, acc);
    }
#pragma unroll
    for (int r = 0; r < 8; ++r) {
      int n = n0 + r + 8 * hi;                   // M dim = n here
      float d = acc[r] + vls[lo * NN + n] * swg[n] + biasg[n * DK + lo];
      qls[lo * NN + n] = d;                      // staged as [y][n]
    }
  }
  __syncthreads();

  float* O = T1 + ((size_t)(b * LL + l) * DM + h * DK) * NN;
  for (int o = tid; o < DK * NN; o += 256) O[o] = qls[o];
}

// ---------------------------------------------------------------------------
// c-conv + attn_w/b + residual(input) + LayerNorm over (D,N), per (b,l).
// All 8 output bands accumulated per n-tile: 8 WMMA per activation fetch,
// slice streamed once per pass instead of 8x.
// ---------------------------------------------------------------------------
__global__ __launch_bounds__(256) void k_conv_ln(
    const float* __restrict__ T1, const float* __restrict__ Xt,
    const float* __restrict__ cW, const float* __restrict__ cb,
    const float* __restrict__ attn_w, const float* __restrict__ attn_b,
    float* __restrict__ T2) {
  __shared__ float Wl[DM * DM];
  __shared__ float cbl[DM];
  __shared__ float red[512];
  __shared__ float stats[2];
  int bl = blockIdx.x, l = bl % LL;
  int tid = threadIdx.x;
  int lane = tid & 31, lo = lane & 15, hi = lane >> 4, wv = tid >> 5;

  for (int i = tid; i < DM * DM; i += 256) Wl[i] = cW[i];
  if (tid < DM) cbl[tid] = cb[tid];
  __syncthreads();

  const float* Sin = T1 + (size_t)bl * DM * NN;
  const float* X0  = Xt + (size_t)bl * DM * NN;
  float* Dst = T2 + (size_t)bl * DM * NN;
  float mean = 0.f, inv = 0.f;

  for (int pass = 0; pass < 2; ++pass) {
    float sum = 0.f, ssq = 0.f;
    for (int nt = wv; nt < 64; nt += 8) {
      int n0 = nt * 16;
      v8f acc[8];
#pragma unroll
      for (int od = 0; od < 8; ++od) acc[od] = (v8f){};
      for (int kk = 0; kk < 32; ++kk) {
        int kr = kk * 4 + hi * 2;
        v2f Bx;
        Bx.x = Sin[(kr + 0) * NN + n0 + lo];
        Bx.y = Sin[(kr + 1) * NN + n0 + lo];
#pragma unroll
        for (int od = 0; od < 8; ++od) {
          v2f A;
          A.x = Wl[(od * 16 + lo) * DM + kr];
          A.y = Wl[(od * 16 + lo) * DM + kr + 1];
          acc[od] = wmma4(A, Bx, acc[od]);
        }
      }
#pragma unroll
      for (int od = 0; od < 8; ++od) {
#pragma unroll
        for (int r = 0; r < 8; ++r) {
          int d = od * 16 + r + 8 * hi, n = n0 + lo;
          float y = fmaxf(acc[od][r] + cbl[d], 0.f);
          size_t ai = ((size_t)d * NN + n) * LL + l;
          float tv = y * attn_w[ai] + attn_b[ai] + y + X0[d * NN + n];
          if (pass == 0) { sum += tv; ssq += tv * tv; }
          else Dst[d * NN + n] = (tv - mean) * inv;
        }
      }
    }
    if (pass == 0) {
      red[tid] = sum; red[256 + tid] = ssq; __syncthreads();
      for (int s = 128; s > 0; s >>= 1) {
        if (tid < s) { red[tid] += red[tid + s]; red[256 + tid] += red[256 + tid + s]; }
        __syncthreads();
      }
      if (tid == 0) {
        float m = red[0] / (float)(DM * NN);
        float var = red[256] / (float)(DM * NN) - m * m;
        stats[0] = m; stats[1] = rsqrtf(var + 1e-5f);
      }
      __syncthreads();
      mean = stats[0]; inv = stats[1];
    }
  }
}

// ---------------------------------------------------------------------------
// GLU gate: g = (g1W@xn + g1b) * sigmoid(g2W@xn + g2b). Per (b,l,od),
// two n-tiles jointly (4 WMMA per shared A fetch).
// ---------------------------------------------------------------------------
__global__ __launch_bounds__(256) void k_glu(
    const float* __restrict__ T2, const float* __restrict__ g1W,
    const float* __restrict__ g1b, const float* __restrict__ g2W,
    const float* __restrict__ g2b, float* __restrict__ T1) {
  __shared__ float W1l[DK * DM];
  __shared__ float W2l[DK * DM];
  __shared__ float b1l[DK], b2l[DK];
  int od = blockIdx.x & 7;
  int bl = blockIdx.x >> 3;
  int tid = threadIdx.x;
  int lane = tid & 31, lo = lane & 15, hi = lane >> 4, wv = tid >> 5;

  for (int i = tid; i < DK * DM; i += 256) {
    int o = i / DM, c = i % DM;
    W1l[i] = g1W[(od * DK + o) * DM + c];
    W2l[i] = g2W[(od * DK + o) * DM + c];
  }
  if (tid < DK) { b1l[tid] = g1b[od * DK + tid]; b2l[tid] = g2b[od * DK + tid]; }
  __syncthreads();

  const float* Sin = T2 + (size_t)bl * DM * NN;
  float* Dst = T1 + (size_t)bl * DM * NN;

  for (int tp = wv; tp < 32; tp += 8) {
    int n0 = tp * 32;
    v8f a10 = {}, a20 = {}, a11 = {}, a21 = {};
    for (int kk = 0; kk < 32; ++kk) {
      int kr = kk * 4 + hi * 2;
      v2f A1, A2, B0, B1;
      A1.x = W1l[lo * DM + kr]; A1.y = W1l[lo * DM + kr + 1];
      A2.x = W2l[lo * DM + kr]; A2.y = W2l[lo * DM + kr + 1];
      B0.x = Sin[(kr + 0) * NN + n0 + lo];
      B0.y = Sin[(kr + 1) * NN + n0 + lo];
      B1.x = Sin[(kr + 0) * NN + n0 + 16 + lo];
      B1.y = Sin[(kr + 1) * NN + n0 + 16 + lo];
      a10 = wmma4(A1, B0, a10);
      a20 = wmma4(A2, B0, a20);
      a11 = wmma4(A1, B1, a11);
      a21 = wmma4(A2, B1, a21);
    }
#pragma unroll
    for (int r = 0; r < 8; ++r) {
      int x = r + 8 * hi;
      int d = od * DK + x;
      float x10 = a10[r] + b1l[x], x20 = a20[r] + b2l[x];
      float x11 = a11[r] + b1l[x], x21 = a21[r] + b2l[x];
      Dst[d * NN + n0 + lo]      = x10 * (1.f / (1.f + __expf(-x20)));
      Dst[d * NN + n0 + 16 + lo] = x11 * (1.f / (1.f + __expf(-x21)));
    }
  }
}

// ---------------------------------------------------------------------------
// g3-conv + residual(xn) + attn_w/b + LayerNorm2, per (b,l) -> U ([b][l][d][n])
// 8 output bands per n-tile, same structure as k_conv_ln.
// ---------------------------------------------------------------------------
__global__ __launch_bounds__(256) void k_final(
    const float* __restrict__ G, const float* __restrict__ XN,
    const float* __restrict__ g3W, const float* __restrict__ g3b,
    const float* __restrict__ attn_w, const float* __restrict__ attn_b,
    float* __restrict__ U) {
  __shared__ float Wl[DM * DM];
  __shared__ float bl3[DM];
  __shared__ float red[512];
  __shared__ float stats[2];
  int bl = blockIdx.x, l = bl % LL;
  int tid = threadIdx.x;
  int lane = tid & 31, lo = lane & 15, hi = lane >> 4, wv = tid >> 5;

  for (int i = tid; i < DM * DM; i += 256) Wl[i] = g3W[i];
  if (tid < DM) bl3[tid] = g3b[tid];
  __syncthreads();

  const float* Sin = G  + (size_t)bl * DM * NN;
  const float* Xn  = XN + (size_t)bl * DM * NN;
  float* Dst = U + (size_t)bl * DM * NN;
  float mean = 0.f, inv = 0.f;

  for (int pass = 0; pass < 2; ++pass) {
    float sum = 0.f, ssq = 0.f;
    for (int nt = wv; nt < 64; nt += 8) {
      int n0 = nt * 16;
      v8f acc[8];
#pragma unroll
      for (int od = 0; od < 8; ++od) acc[od] = (v8f){};
      for (int kk = 0; kk < 32; ++kk) {
        int kr = kk * 4 + hi * 2;
        v2f Bx;
        Bx.x = Sin[(kr + 0) * NN + n0 + lo];
        Bx.y = Sin[(kr + 1) * NN + n0 + lo];
#pragma unroll
        for (int od = 0; od < 8; ++od) {
          v2f A;
          A.x = Wl[(od * 16 + lo) * DM + kr];
          A.y = Wl[(od * 16 + lo) * DM + kr + 1];
          acc[od] = wmma4(A, Bx, acc[od]);
        }
      }
#pragma unroll
      for (int od = 0; od < 8; ++od) {
#pragma unroll
        for (int r = 0; r < 8; ++r) {
          int d = od * 16 + r + 8 * hi, n = n0 + lo;
          float x3 = acc[od][r] + bl3[d];
          float u0 = x3 + Xn[d * NN + n];
          size_t ai = ((size_t)d * NN + n) * LL + l;
          float tv = u0 * attn_w[ai] + attn_b[ai] + u0;
          if (pass == 0) { sum += tv; ssq += tv * tv; }
          else Dst[d * NN + n] = (tv - mean) * inv;
        }
      }
    }
    if (pass == 0) {
      red[tid] = sum; red[256 + tid] = ssq; __syncthreads();
      for (int s = 128; s > 0; s >>= 1) {
        if (tid < s) { red[tid] += red[tid + s]; red[256 + tid] += red[256 + tid + s]; }
        __syncthreads();
      }
      if (tid == 0) {
        float m = red[0] / (float)(DM * NN);
        float var = red[256] / (float)(DM * NN) - m * m;
        stats[0] = m; stats[1] = rsqrtf(var + 1e-5f);
      }
      __syncthreads();
      mean = stats[0]; inv = stats[1];
    }
  }
}

// ---------------------------------------------------------------------------
extern "C" void kernel_launch(void* const* d_in, const int* in_sizes, int n_in,
                              void* d_out, int out_size, void* d_ws, size_t ws_size,
                              hipStream_t stream) {
  (void)in_sizes; (void)n_in; (void)out_size; (void)ws_size;
  const float* input  = (const float*)d_in[0];
  const float* qW     = (const float*)d_in[1];
  const float* qb     = (const float*)d_in[2];
  const float* vW     = (const float*)d_in[3];
  const float* vb     = (const float*)d_in[4];
  const float* cW     = (const float*)d_in[5];
  const float* cb     = (const float*)d_in[6];
  const float* memory = (const float*)d_in[7];
  const float* attn_w = (const float*)d_in[8];
  const float* attn_b = (const float*)d_in[9];
  const float* nv1    = (const float*)d_in[10];
  const float* nv2    = (const float*)d_in[11];
  const float* w_pool = (const float*)d_in[12];
  const float* b_pool = (const float*)d_in[13];
  const float* g1W    = (const float*)d_in[14];
  const float* g1b    = (const float*)d_in[15];
  const float* g2W    = (const float*)d_in[16];
  const float* g2b    = (const float*)d_in[17];
  const float* g3W    = (const float*)d_in[18];
  const float* g3b    = (const float*)d_in[19];
  float* out = (float*)d_out;

  float* ws = (float*)d_ws;
  const size_t TEN = (size_t)BB * DM * NN * LL;      // 25,165,824 floats
  float* Xt    = ws;
  float* T1    = Xt + TEN;
  float* T2    = T1 + TEN;
  float* keysm = T2 + TEN;                           // 8*12*1024*16
  float* biasg = keysm + (size_t)HD * LL * NN * DK;  // 1024*16
  float* swg   = biasg + (size_t)NN * DK;            // 1024

  k_transpose_in<<<BB * DM, 256, 0, stream>>>(input, Xt);
  k_key<<<(HD * LL * NN) / 256, 256, 0, stream>>>(memory, keysm);
  k_apt<<<NN, 256, 0, stream>>>(nv1, nv2, w_pool, b_pool, biasg, swg);
  k_attn<<<BB * HD * LL, 256, 0, stream>>>(Xt, qW, qb, vW, vb, keysm, biasg, swg, T1);
  k_conv_ln<<<BB * LL, 256, 0, stream>>>(T1, Xt, cW, cb, attn_w, attn_b, T2);
  k_glu<<<BB * LL * 8, 256, 0, stream>>>(T2, g1W, g1b, g2W, g2b, T1);
  k_final<<<BB * LL, 256, 0, stream>>>(T1, T2, g3W, g3b, attn_w, attn_b, Xt);
  k_transpose_out<<<BB * DM, 256, 0, stream>>>(Xt, out);
}